// GlobalStateCore_13812614824434
// MI455X (gfx1250) — compile-verified
//
#include <hip/hip_runtime.h>
#include <math.h>

// Sizes from the reference
#define T_DIM 256
#define B_DIM 512
#define EMBED 256
#define ACT   7
#define MEM   200
#define STOCH 30
#define HID   200
#define EA_IN (EMBED + ACT)   // 263
#define G3    (3 * MEM)       // 600

// Padded (tile-aligned) shapes kept in workspace
#define KEA_P 264   // 263 -> 33 dual k-steps of 8
#define HID_P 208   // 200 -> 13 col tiles
#define G3_P  608   // 600 -> 38 col tiles
#define OUT_P 64    // 60  -> 4 col tiles

// LDS row strides: even (8B-aligned v2f reads) and (stride mod 64) generates
// 16 distinct banks across lanes 0-15 -> conflict-free ds_load_b64.
#define XS_LD 266   // 266 % 64 = 10
#define ES_LD 210   // 210 % 64 = 18
#define HS_LD 210
#define MS_LD 210
#define GH_LD 610

typedef __attribute__((ext_vector_type(2))) float v2f;
typedef __attribute__((ext_vector_type(8))) float v8f;

// Full-precision CDNA5 WMMA: D(16x16,f32) = A(16x4,f32) x B(4x16,f32) + C
__device__ __forceinline__ v8f wmma_f32_16x16x4(v2f a, v2f b, v8f c) {
  return __builtin_amdgcn_wmma_f32_16x16x4_f32(false, a, false, b, (short)0, c,
                                               false, false);
}

__device__ __forceinline__ float sigmoidf_(float x) { return 1.0f / (1.0f + expf(-x)); }
__device__ __forceinline__ float eluf_(float x)     { return x > 0.0f ? x : expm1f(x); }
__device__ __forceinline__ float softplusf_(float x){ return x > 20.0f ? x : log1pf(expf(x)); }

// Unconditional B-fragment load from a padded, row-major [K][ldb] weight panel.
__device__ __forceinline__ v2f load_b_frag(const float* __restrict__ W, int ldb,
                                           int k0, int n, int half) {
  const int kx = k0 + 2 * half;
  v2f b;
  b.x = W[kx * ldb + n];           // lanes 0-15: 64B contiguous (coalesced)
  b.y = W[(kx + 1) * ldb + n];
  return b;
}

// Dual-accumulator K-loop body: two independent WMMA dependency chains.
// Arow must point at the lane's LDS row; K must be a multiple of 8.
#define GEMM_K_LOOP(K, Arow, Wp, ldb)                                          \
  v8f acc0 = {}, acc1 = {};                                                    \
  for (int k0 = 0; k0 < (K); k0 += 8) {                                        \
    v2f a0 = *(const v2f*)&(Arow)[k0 + 2 * half];                              \
    v2f b0 = load_b_frag((Wp), (ldb), k0, n, half);                            \
    acc0 = wmma_f32_16x16x4(a0, b0, acc0);                                     \
    v2f a1 = *(const v2f*)&(Arow)[k0 + 4 + 2 * half];                          \
    v2f b1 = load_b_frag((Wp), (ldb), k0 + 4, n, half);                        \
    acc1 = wmma_f32_16x16x4(a1, b1, acc1);                                     \
  }                                                                            \
  v8f acc = acc0 + acc1;

// Zero-padded 2D copy: dst[drows][dcols] <- src[srows][scols]
__global__ void pad2d_kernel(const float* __restrict__ src, float* __restrict__ dst,
                             int srows, int scols, int drows, int dcols) {
  int i = blockIdx.x * blockDim.x + threadIdx.x;
  if (i < drows * dcols) {
    int r = i / dcols, c = i % dcols;
    dst[i] = (r < srows && c < scols) ? src[r * scols + c] : 0.0f;
  }
}

// ---------------------------------------------------------------------------
// Kernel 1 (fully parallel over T*B rows):
//   gi_all[T*B,608] = elu([embed|action] @ W_ea + b_ea) @ W_ih + b_ih
// ---------------------------------------------------------------------------
__global__ void __launch_bounds__(256)
gi_precompute_kernel(const float* __restrict__ embed, const float* __restrict__ action,
                     const float* __restrict__ Wea_p, const float* __restrict__ bea_p,
                     const float* __restrict__ Wih_p, const float* __restrict__ bih_p,
                     float* __restrict__ gi_all) {
  __shared__ __align__(16) float Xs[16][XS_LD];
  __shared__ __align__(16) float Es[16][ES_LD];

  const int tid  = threadIdx.x;
  const int wave = tid >> 5;
  const int lane = tid & 31;
  const int half = lane >> 4;
  const int l    = lane & 15;
  const long row0 = (long)blockIdx.x * 16;   // global row in [0, T*B)

  // Stage X = [embed | action | 0] for 16 rows
  for (int i = tid; i < 16 * KEA_P; i += 256) {
    int r = i / KEA_P, c = i % KEA_P;
    long gr = row0 + r;
    float v;
    if (c < EMBED)      v = embed[gr * EMBED + c];
    else if (c < EA_IN) v = action[gr * ACT + (c - EMBED)];
    else                v = 0.0f;
    Xs[r][c] = v;
  }
  __syncthreads();

  // GEMM1: ea = elu(X @ W_ea + b_ea); 13 col tiles; K = 264 (padded)
  for (int jt = wave; jt < HID_P / 16; jt += 8) {
    const int n = jt * 16 + l;
    GEMM_K_LOOP(KEA_P, Xs[l], Wea_p, HID_P);
    const float bias = bea_p[n];               // padded: 0 beyond 200
    for (int i = 0; i < 8; ++i)
      Es[i + 8 * half][n] = eluf_(acc[i] + bias);
  }
  __syncthreads();

  // GEMM2: gi = ea @ W_ih + b_ih; 38 col tiles; K = 200
  float* gi_row0 = gi_all + row0 * G3_P;
  for (int jt = wave; jt < G3_P / 16; jt += 8) {
    const int n = jt * 16 + l;
    GEMM_K_LOOP(HID, Es[l], Wih_p, G3_P);
    const float bias = bih_p[n];
    for (int i = 0; i < 8; ++i)
      gi_row0[(long)(i + 8 * half) * G3_P + n] = acc[i] + bias;
  }
}

// ---------------------------------------------------------------------------
// Kernel 2 (sequential, one launch per timestep):
//   h = h_prev * !reset[t];  gh = h @ W_hh;  GRU gates;  states[t] = h_new
// ---------------------------------------------------------------------------
__global__ void __launch_bounds__(256)
gru_step_kernel(int t, const float* __restrict__ h_in,
                const unsigned char* __restrict__ reset,   // bool tensor, 1B/elem
                const float* __restrict__ Whh_p, const float* __restrict__ b_hh,
                const float* __restrict__ gi_all,
                float* __restrict__ states) {
  __shared__ __align__(16) float Hs[16][HS_LD];
  __shared__ __align__(16) float Gh[16][GH_LD];

  const int tid  = threadIdx.x;
  const int wave = tid >> 5;
  const int lane = tid & 31;
  const int half = lane >> 4;
  const int l    = lane & 15;
  const int b0   = blockIdx.x * 16;

  // Load h_prev, apply reset mask (also zero the K-pad columns)
  for (int i = tid; i < 16 * HS_LD; i += 256) {
    int r = i / HS_LD, m = i % HS_LD;
    float v = 0.0f;
    if (m < MEM) {
      v = h_in[(long)(b0 + r) * MEM + m];
      if (reset[(long)t * B_DIM + b0 + r]) v = 0.0f;
    }
    Hs[r][m] = v;
  }
  __syncthreads();

  // gh = h @ W_hh (pre-bias); 38 col tiles; K = 200
  for (int jt = wave; jt < G3_P / 16; jt += 8) {
    const int n = jt * 16 + l;
    GEMM_K_LOOP(MEM, Hs[l], Whh_p, G3_P);
    for (int i = 0; i < 8; ++i)
      Gh[i + 8 * half][n] = acc[i];
  }
  __syncthreads();

  // GRU gates + state update (torch GRUCell semantics)
  float* st = states + (long)t * B_DIM * MEM;
  for (int i = tid; i < 16 * MEM; i += 256) {
    int r = i / MEM, m = i % MEM;
    int b = b0 + r;
    const float* gi = gi_all + ((long)t * B_DIM + b) * G3_P;
    float g_r = gi[m]           + Gh[r][m]           + b_hh[m];
    float g_z = gi[MEM + m]     + Gh[r][MEM + m]     + b_hh[MEM + m];
    float gin = gi[2 * MEM + m];
    float ghn = Gh[r][2 * MEM + m] + b_hh[2 * MEM + m];
    float rg = sigmoidf_(g_r);
    float zg = sigmoidf_(g_z);
    float ng = tanhf(gin + rg * ghn);
    float hp = Hs[r][m];
    st[(long)b * MEM + m] = (1.0f - zg) * ng + zg * hp;
  }
}

// ---------------------------------------------------------------------------
// Kernel 3 (fully parallel over T*B rows):
//   mid = elu(states @ Wp1 + bp1); out = mid @ Wp2 + bp2
//   posts[:, :30] = mean; posts[:, 30:] = softplus(raw) + 0.1
// ---------------------------------------------------------------------------
__global__ void __launch_bounds__(256)
post_mlp_kernel(const float* __restrict__ states,
                const float* __restrict__ Wp1_p, const float* __restrict__ bp1_p,
                const float* __restrict__ Wp2_p, const float* __restrict__ bp2_p,
                float* __restrict__ posts) {
  __shared__ __align__(16) float Hs[16][HS_LD];
  __shared__ __align__(16) float Ms[16][MS_LD];

  const int tid  = threadIdx.x;
  const int wave = tid >> 5;
  const int lane = tid & 31;
  const int half = lane >> 4;
  const int l    = lane & 15;
  const long row0 = (long)blockIdx.x * 16;

  for (int i = tid; i < 16 * HS_LD; i += 256) {
    int r = i / HS_LD, m = i % HS_LD;
    Hs[r][m] = (m < MEM) ? states[(row0 + r) * MEM + m] : 0.0f;
  }
  __syncthreads();

  // GEMM1: mid = elu(h @ Wp1 + bp1); 13 col tiles; K = 200
  for (int jt = wave; jt < HID_P / 16; jt += 8) {
    const int n = jt * 16 + l;
    GEMM_K_LOOP(MEM, Hs[l], Wp1_p, HID_P);
    const float bias = bp1_p[n];
    for (int i = 0; i < 8; ++i)
      Ms[i + 8 * half][n] = eluf_(acc[i] + bias);
  }
  __syncthreads();

  // GEMM2: out = mid @ Wp2 + bp2; 4 col tiles cover 60; K = 200
  for (int jt = wave; jt < OUT_P / 16; jt += 8) {
    const int n = jt * 16 + l;
    GEMM_K_LOOP(MEM, Ms[l], Wp2_p, OUT_P);
    if (n < 2 * STOCH) {
      const float bias = bp2_p[n];
      for (int i = 0; i < 8; ++i) {
        int m = i + 8 * half;
        float v = acc[i] + bias;
        if (n >= STOCH) v = softplusf_(v) + 0.1f;   // MIN_STD
        posts[(row0 + m) * (2 * STOCH) + n] = v;
      }
    }
  }
}

// sample = mean + std * noise from posts[T-1]
__global__ void sample_kernel(const float* __restrict__ posts,
                              const float* __restrict__ noise,
                              float* __restrict__ sample) {
  int i = blockIdx.x * blockDim.x + threadIdx.x;
  if (i < B_DIM * STOCH) {
    int b = i / STOCH, s = i % STOCH;
    const float* p = posts + ((long)(T_DIM - 1) * B_DIM + b) * (2 * STOCH);
    sample[i] = p[s] + p[STOCH + s] * noise[i];
  }
}

extern "C" void kernel_launch(void* const* d_in, const int* in_sizes, int n_in,
                              void* d_out, int out_size, void* d_ws, size_t ws_size,
                              hipStream_t stream) {
  (void)in_sizes; (void)n_in; (void)out_size; (void)ws_size;

  const float* embed          = (const float*)d_in[0];
  const float* action         = (const float*)d_in[1];
  const unsigned char* reset  = (const unsigned char*)d_in[2];  // bool -> 1 byte
  const float* in_state       = (const float*)d_in[3];
  const float* noise          = (const float*)d_in[4];
  const float* W_ea = (const float*)d_in[5];
  const float* b_ea = (const float*)d_in[6];
  const float* W_ih = (const float*)d_in[7];
  const float* b_ih = (const float*)d_in[8];
  const float* W_hh = (const float*)d_in[9];
  const float* b_hh = (const float*)d_in[10];
  const float* Wp1  = (const float*)d_in[11];
  const float* bp1  = (const float*)d_in[12];
  const float* Wp2  = (const float*)d_in[13];
  const float* bp2  = (const float*)d_in[14];

  // Output layout: sample [B,30] | states [T,B,200] | posts [T,B,60]
  float* out_sample = (float*)d_out;
  float* states = out_sample + (size_t)B_DIM * STOCH;
  float* posts  = states + (size_t)T_DIM * B_DIM * MEM;

  // Workspace layout (floats)
  float* ws = (float*)d_ws;
  float* gi_all = ws;                                  // [T*B, 608]
  size_t off = (size_t)T_DIM * B_DIM * G3_P;
  float* Wea_p = ws + off;  off += (size_t)KEA_P * HID_P;
  float* Wih_p = ws + off;  off += (size_t)HID * G3_P;
  float* Whh_p = ws + off;  off += (size_t)HID * G3_P;
  float* Wp1_p = ws + off;  off += (size_t)HID * HID_P;
  float* Wp2_p = ws + off;  off += (size_t)HID * OUT_P;
  float* bea_p = ws + off;  off += HID_P;
  float* bih_p = ws + off;  off += G3_P;
  float* bp1_p = ws + off;  off += HID_P;
  float* bp2_p = ws + off;  off += OUT_P;

  // Phase 0: zero-pad weights/biases into tile-aligned panels (removes all
  // per-lane guards from the WMMA inner loops).
  #define PAD(src, dst, sr, sc, dr, dc)                                        \
    pad2d_kernel<<<((dr) * (dc) + 255) / 256, 256, 0, stream>>>(src, dst, sr, sc, dr, dc)
  PAD(W_ea, Wea_p, EA_IN, HID, KEA_P, HID_P);
  PAD(W_ih, Wih_p, HID, G3, HID, G3_P);
  PAD(W_hh, Whh_p, HID, G3, HID, G3_P);
  PAD(Wp1,  Wp1_p, HID, HID, HID, HID_P);
  PAD(Wp2,  Wp2_p, HID, 2 * STOCH, HID, OUT_P);
  PAD(b_ea, bea_p, 1, HID, 1, HID_P);
  PAD(b_ih, bih_p, 1, G3, 1, G3_P);
  PAD(bp1,  bp1_p, 1, HID, 1, HID_P);
  PAD(bp2,  bp2_p, 1, 2 * STOCH, 1, OUT_P);
  #undef PAD

  const int nrow_tiles = (T_DIM * B_DIM) / 16;   // 8192 workgroups

  // Phase 1: parallel precompute of input-side GRU gates (45 GFLOP, full chip)
  gi_precompute_kernel<<<nrow_tiles, 256, 0, stream>>>(
      embed, action, Wea_p, bea_p, Wih_p, bih_p, gi_all);

  // Phase 2: 256 dependent GRU steps (latency-bound; W_hh stays L2-resident)
  for (int t = 0; t < T_DIM; ++t) {
    const float* h_in = (t == 0) ? in_state
                                 : (states + (size_t)(t - 1) * B_DIM * MEM);
    gru_step_kernel<<<B_DIM / 16, 256, 0, stream>>>(
        t, h_in, reset, Whh_p, b_hh, gi_all, states);
  }

  // Phase 3: parallel post-MLP over all stored states (13.6 GFLOP)
  post_mlp_kernel<<<nrow_tiles, 256, 0, stream>>>(
      states, Wp1_p, bp1_p, Wp2_p, bp2_p, posts);

  // Phase 4: rsample from posts[T-1]
  sample_kernel<<<(B_DIM * STOCH + 255) / 256, 256, 0, stream>>>(
      posts, noise, out_sample);
}